// Simulator_46943992546089
// MI455X (gfx1250) — compile-verified
//
#include <hip/hip_runtime.h>

#define HH 256
#define WW 256
#define NPIX (HH*WW)
#define KLEN 65
#define PADR 32
#define CSTRIDE 112

typedef __attribute__((ext_vector_type(2))) float v2f;
typedef __attribute__((ext_vector_type(8))) float v8f;

__device__ __forceinline__ int refl(int i, int n) {
  if (i < 0) i = -i;
  if (i >= n) i = 2*n - 2 - i;
  return i;
}

__device__ __forceinline__ v8f vzero8() {
  v8f z = {0.f,0.f,0.f,0.f,0.f,0.f,0.f,0.f};
  return z;
}

// ------------- Kernel 1: bilinear grid-sample, border padding, align_corners=False
__global__ void __launch_bounds__(256) k_tilt(
    const float* __restrict__ img, const float* __restrict__ zern,
    float* __restrict__ tilt)
{
  int p = blockIdx.x*256 + threadIdx.x;
  if (p >= NPIX) return;
  int h = p >> 8, w = p & 255;
  float px = zern[p*35+0], py = zern[p*35+1];
  // ix = (x+px)*W/(W-1) - 0.5  (derived from the reference flow formula)
  float ix = ((float)w + px) * ((float)WW/(float)(WW-1)) - 0.5f;
  float iy = ((float)h + py) * ((float)HH/(float)(HH-1)) - 0.5f;
  ix = fminf(fmaxf(ix, 0.f), (float)(WW-1));
  iy = fminf(fmaxf(iy, 0.f), (float)(HH-1));
  float x0f = floorf(ix), y0f = floorf(iy);
  float wx = ix - x0f, wy = iy - y0f;
  int x0 = min(max((int)x0f, 0), WW-1);
  int y0 = min(max((int)y0f, 0), HH-1);
  int x1 = min(x0+1, WW-1), y1 = min(y0+1, HH-1);
  #pragma unroll
  for (int c = 0; c < 3; ++c) {
    const float* im = img + c*NPIX;
    float t00 = im[y0*WW+x0], t01 = im[y0*WW+x1];
    float t10 = im[y1*WW+x0], t11 = im[y1*WW+x1];
    float top = t00 + (t01 - t00)*wx;
    float bot = t10 + (t11 - t10)*wx;
    tilt[c*NPIX+p] = top + (bot - top)*wy;
  }
}

// ------------- Kernel 2: fused per-pixel MLP 33->200->200->100 via WMMA fp32
// B fragment: B[k][n] = Wrow_major[n][k]; lanes carry N, VGPRs carry K (mirrors A).
__device__ __forceinline__ v2f wmma_ld_b(const float* __restrict__ Wm,
                                         int Nn, int Kk, int nt, int kt, int lane) {
  int n = nt*16 + (lane & 15);
  int k0 = kt*4 + ((lane & 16) ? 2 : 0);
  v2f b; b.x = 0.f; b.y = 0.f;
  if (n < Nn) {
    const float* r = Wm + n*Kk;
    if (k0   < Kk) b.x = r[k0];
    if (k0+1 < Kk) b.y = r[k0+1];
  }
  return b;
}

__global__ void __launch_bounds__(128) k_mlp(
    const float* __restrict__ zern,
    const float* __restrict__ w1, const float* __restrict__ b1,
    const float* __restrict__ w2, const float* __restrict__ b2,
    const float* __restrict__ w3, const float* __restrict__ b3,
    float* __restrict__ coef)
{
  __shared__ float hbuf[4][16][208];          // 53,248 B: per-wave hidden tile
  const int lane  = threadIdx.x & 31;
  const int wv    = threadIdx.x >> 5;
  const int mbase = (blockIdx.x*4 + wv) * 16; // 16-pixel M tile per wave
  float (*hb)[208] = hbuf[wv];
  const int mrow  = lane & 15;
  const int koff  = (lane & 16) ? 2 : 0;
  const int rbase = (lane & 16) ? 8 : 0;

  // ---- layer 1: K=33 (pad 36), N=200 (pad 208), leaky relu ----
  v2f afr[9];
  #pragma unroll
  for (int kt = 0; kt < 9; ++kt) {
    int k0 = kt*4 + koff;
    const float* zr = zern + (size_t)(mbase + mrow)*35 + 2;
    v2f a;
    a.x = (k0   < 33) ? zr[k0]   : 0.f;
    a.y = (k0+1 < 33) ? zr[k0+1] : 0.f;
    afr[kt] = a;
  }
  for (int nt = 0; nt < 13; ++nt) {
    v8f acc = vzero8();
    #pragma unroll
    for (int kt = 0; kt < 9; ++kt) {
      v2f b = wmma_ld_b(w1, 200, 33, nt, kt, lane);
      acc = __builtin_amdgcn_wmma_f32_16x16x4_f32(false, afr[kt], false, b,
                                                  (short)0, acc, false, false);
    }
    int col = nt*16 + mrow;
    float bias = (col < 200) ? b1[col] : 0.f;
    #pragma unroll
    for (int r = 0; r < 8; ++r) {
      float v = acc[r] + bias;
      v = (v > 0.f) ? v : 0.01f*v;
      hb[rbase + r][col] = v;        // cols 200..207 become exactly 0
    }
  }

  // ---- layer 2: K=200, N=200, leaky relu (accumulate all tiles, then rewrite LDS) ----
  v8f acc2[13];
  for (int nt = 0; nt < 13; ++nt) {
    v8f acc = vzero8();
    for (int kt = 0; kt < 50; ++kt) {
      int k0 = kt*4 + koff;
      v2f a; a.x = hb[mrow][k0]; a.y = hb[mrow][k0+1];
      v2f b = wmma_ld_b(w2, 200, 200, nt, kt, lane);
      acc = __builtin_amdgcn_wmma_f32_16x16x4_f32(false, a, false, b,
                                                  (short)0, acc, false, false);
    }
    acc2[nt] = acc;
  }
  for (int nt = 0; nt < 13; ++nt) {
    int col = nt*16 + mrow;
    float bias = (col < 200) ? b2[col] : 0.f;
    #pragma unroll
    for (int r = 0; r < 8; ++r) {
      float v = acc2[nt][r] + bias;
      v = (v > 0.f) ? v : 0.01f*v;
      hb[rbase + r][col] = v;        // DS ops are in-order per wave: safe
    }
  }

  // ---- layer 3: K=200, N=100 (pad 112), linear; write coef[pixel][112] ----
  for (int nt = 0; nt < 7; ++nt) {
    v8f acc = vzero8();
    for (int kt = 0; kt < 50; ++kt) {
      int k0 = kt*4 + koff;
      v2f a; a.x = hb[mrow][k0]; a.y = hb[mrow][k0+1];
      v2f b = wmma_ld_b(w3, 100, 200, nt, kt, lane);
      acc = __builtin_amdgcn_wmma_f32_16x16x4_f32(false, a, false, b,
                                                  (short)0, acc, false, false);
    }
    int col = nt*16 + mrow;
    float bias = (col < 100) ? b3[col] : 0.f;
    #pragma unroll
    for (int r = 0; r < 8; ++r) {
      coef[(size_t)(mbase + rbase + r)*CSTRIDE + col] = acc[r] + bias;
    }
  }
}

// ------------- Kernel 3: vertical pass, folded over 'a'
// Vc[c][b] = sum_t (sum_a L_a[t] coef_ab) * tilt_c      (30 planes)
// Vmu[c][m] = sum_t M_m[t] * tilt_c                      (30 planes)
// Db[b]     = sum_t (sum_a L_a[t] coef_ab)               (10 planes)
__global__ void __launch_bounds__(256) k_vert(
    const float* __restrict__ coef, const float* __restrict__ tilt,
    const float* __restrict__ basis_left, const float* __restrict__ mu,
    float* __restrict__ vbuf)
{
  __shared__ float sc[128][100];     // 51,200 B
  __shared__ float st[3][128];       //  1,536 B
  __shared__ float sbl[KLEN*10];     //  2,600 B
  __shared__ float smu[KLEN*10];     //  2,600 B
  const int w  = blockIdx.x;
  const int h0 = blockIdx.y * 64;
  const int tid = threadIdx.x;

  for (int idx = tid; idx < 128*100; idx += 256) {
    int i = idx / 100, ch = idx % 100;
    int hr = refl(h0 - PADR + i, HH);
    sc[i][ch] = coef[(size_t)(hr*WW + w)*CSTRIDE + ch];
  }
  for (int idx = tid; idx < 3*128; idx += 256) {
    int c = idx / 128, i = idx % 128;
    int hr = refl(h0 - PADR + i, HH);
    st[c][i] = tilt[c*NPIX + hr*WW + w];
  }
  for (int idx = tid; idx < KLEN*10; idx += 256) {
    sbl[idx] = basis_left[idx];
    smu[idx] = mu[idx];
  }
  __syncthreads();

  for (int task = tid; task < 640; task += 256) {
    int h = task / 10, b = task % 10;
    float a0=0.f, a1=0.f, a2=0.f, ad=0.f, m0=0.f, m1=0.f, m2=0.f;
    for (int t = 0; t < KLEN; ++t) {
      const float* cr = &sc[h + t][0];
      const float* lb = &sbl[t*10];
      float inner = 0.f;
      #pragma unroll
      for (int a = 0; a < 10; ++a) inner += lb[a] * cr[a*10 + b];
      float t0 = st[0][h+t], t1 = st[1][h+t], t2 = st[2][h+t];
      ad += inner;
      a0 += inner*t0; a1 += inner*t1; a2 += inner*t2;
      float mv = smu[t*10 + b];
      m0 += mv*t0; m1 += mv*t1; m2 += mv*t2;
    }
    int o = (h0 + h)*WW + w;
    vbuf[(      b)*NPIX + o] = a0;
    vbuf[(10 +  b)*NPIX + o] = a1;
    vbuf[(20 +  b)*NPIX + o] = a2;
    vbuf[(30 +  b)*NPIX + o] = m0;
    vbuf[(40 +  b)*NPIX + o] = m1;
    vbuf[(50 +  b)*NPIX + o] = m2;
    vbuf[(60 +  b)*NPIX + o] = ad;
  }
}

// ------------- Kernel 4: horizontal pass + mu-constant denominator + divide
__global__ void __launch_bounds__(128) k_horz(
    const float* __restrict__ vbuf,
    const float* __restrict__ basis_right, const float* __restrict__ mu,
    float* __restrict__ out)
{
  __shared__ float sv[70][192];      // 53,760 B
  __shared__ float sbr[KLEN*10];
  __shared__ float smu[KLEN*10];
  const int h  = blockIdx.x;
  const int w0 = blockIdx.y * 128;
  const int tid = threadIdx.x;

  for (int idx = tid; idx < 70*192; idx += 128) {
    int pl = idx / 192, i = idx % 192;
    int wr = refl(w0 - PADR + i, WW);
    sv[pl][i] = vbuf[(size_t)pl*NPIX + h*WW + wr];
  }
  for (int idx = tid; idx < KLEN*10; idx += 128) {
    sbr[idx] = basis_right[idx];
    smu[idx] = mu[idx];
  }
  __syncthreads();

  // denominator mu part: sum_m (sum_t mu[t][m])^2  (constant)
  float denc = 0.f;
  #pragma unroll
  for (int m = 0; m < 10; ++m) {
    float s = 0.f;
    for (int t = 0; t < KLEN; ++t) s += smu[t*10 + m];
    denc += s*s;
  }

  const int wl = tid;
  float n0=0.f, n1=0.f, n2=0.f, dd=0.f;
  for (int s = 0; s < KLEN; ++s) {
    int wi = wl + s;
    const float* rb = &sbr[s*10];
    const float* mm = &smu[s*10];
    #pragma unroll
    for (int b = 0; b < 10; ++b) {
      float r = rb[b];
      n0 += r * sv[b][wi];
      n1 += r * sv[10 + b][wi];
      n2 += r * sv[20 + b][wi];
      dd += r * sv[60 + b][wi];
      float mv = mm[b];
      n0 += mv * sv[30 + b][wi];
      n1 += mv * sv[40 + b][wi];
      n2 += mv * sv[50 + b][wi];
    }
  }
  dd += denc;
  int o = h*WW + w0 + wl;
  out[0*NPIX + o] = n0/dd;
  out[1*NPIX + o] = n1/dd;
  out[2*NPIX + o] = n2/dd;
}

extern "C" void kernel_launch(void* const* d_in, const int* in_sizes, int n_in,
                              void* d_out, int out_size, void* d_ws, size_t ws_size,
                              hipStream_t stream) {
  const float* img  = (const float*)d_in[0];
  const float* zern = (const float*)d_in[1];
  const float* w1   = (const float*)d_in[2];
  const float* b1   = (const float*)d_in[3];
  const float* w2   = (const float*)d_in[4];
  const float* b2   = (const float*)d_in[5];
  const float* w3   = (const float*)d_in[6];
  const float* b3   = (const float*)d_in[7];
  const float* bl   = (const float*)d_in[8];
  const float* br   = (const float*)d_in[9];
  const float* muw  = (const float*)d_in[10];
  // d_in[11] = ksize (65), hardcoded
  float* out = (float*)d_out;

  float* tilt = (float*)d_ws;                       // 3*65536 floats
  float* coef = tilt + 3*NPIX;                      // 65536*112 floats
  float* vbuf = coef + (size_t)NPIX*CSTRIDE;        // 70*65536 floats (~48.5 MB total)

  k_tilt<<<NPIX/256, 256, 0, stream>>>(img, zern, tilt);
  k_mlp <<<NPIX/64, 128, 0, stream>>>(zern, w1, b1, w2, b2, w3, b3, coef);
  k_vert<<<dim3(WW, 4), 256, 0, stream>>>(coef, tilt, bl, muw, vbuf);
  k_horz<<<dim3(HH, 2), 128, 0, stream>>>(vbuf, br, muw, out);
}